// HungarianMatcher_18528488915545
// MI455X (gfx1250) — compile-verified
//
#include <hip/hip_runtime.h>
#include <cstdint>

#define NCLS 20
#define QPC  27
#define NQ   540
#define INF_F 3.402823466e38f

typedef int vi4 __attribute__((vector_size(16)));

// ---------------- CDNA5 async global->LDS staging helpers ----------------
#if defined(__HIP_DEVICE_COMPILE__)
#if __has_builtin(__builtin_amdgcn_global_load_async_to_lds_b128) && \
    __has_builtin(__builtin_amdgcn_global_load_async_to_lds_b32)
#define ASYNC_BUILTIN 1
#endif
#endif

__device__ __forceinline__ void wait_async_zero() {
#if defined(__HIP_DEVICE_COMPILE__)
#if __has_builtin(__builtin_amdgcn_s_wait_asynccnt)
  __builtin_amdgcn_s_wait_asynccnt(0);
#else
  asm volatile("s_wait_asynccnt 0x0" ::: "memory");
#endif
#endif
}

__device__ __forceinline__ void async_g2l_b128(const void* g, void* l) {
#if defined(ASYNC_BUILTIN)
  __builtin_amdgcn_global_load_async_to_lds_b128(
      (__attribute__((address_space(1))) vi4*)g,
      (__attribute__((address_space(3))) vi4*)l, 0, 0);
#elif defined(__HIP_DEVICE_COMPILE__)
  unsigned loff = (unsigned)(uintptr_t)(__attribute__((address_space(3))) void*)l;
  unsigned long long ga = (unsigned long long)(uintptr_t)g;
  asm volatile("global_load_async_to_lds_b128 %0, %1, off"
               :: "v"(loff), "v"(ga) : "memory");
#else
  (void)g; (void)l;
#endif
}

__device__ __forceinline__ void async_g2l_b32(const void* g, void* l) {
#if defined(ASYNC_BUILTIN)
  __builtin_amdgcn_global_load_async_to_lds_b32(
      (__attribute__((address_space(1))) int*)g,
      (__attribute__((address_space(3))) int*)l, 0, 0);
#elif defined(__HIP_DEVICE_COMPILE__)
  unsigned loff = (unsigned)(uintptr_t)(__attribute__((address_space(3))) void*)l;
  unsigned long long ga = (unsigned long long)(uintptr_t)g;
  asm volatile("global_load_async_to_lds_b32 %0, %1, off"
               :: "v"(loff), "v"(ga) : "memory");
#else
  (void)g; (void)l;
#endif
}

// ---------------- main kernel: 1 block per batch item, 20 waves ----------
__launch_bounds__(640)
__global__ void matcher_kernel(const float* __restrict__ logits,
                               const float* __restrict__ anchors,
                               const float* __restrict__ tgtb,
                               const int*   __restrict__ tgtp,
                               const int*   __restrict__ ntq,
                               int*   __restrict__ matches,
                               float* __restrict__ soft)
{
  __shared__ __align__(16) float s_tgt[NCLS * 6];   // 480 B
  __shared__ __align__(16) int   s_prs[NCLS];       // 80 B
  __shared__ float s_sv[NQ];                        // soft_vals candidates
  __shared__ unsigned s_mask;

  const int b    = blockIdx.x;
  const int tid  = threadIdx.x;
  const int lane = tid & 31;
  const int wid  = tid >> 5;

  // ---- stage tgt_boxes[b] (30 x 16B) and tgt_present[b] (20 x 4B) to LDS
  if (wid == 0) {
    if (lane < 30)
      async_g2l_b128(tgtb + (size_t)b * (NCLS * 6) + lane * 4, &s_tgt[lane * 4]);
    if (lane < NCLS)
      async_g2l_b32(tgtp + (size_t)b * NCLS + lane, &s_prs[lane]);
    wait_async_zero();
  }
  __syncthreads();

  if (tid == 0) {
    unsigned m = 0;
#pragma unroll
    for (int i = 0; i < NCLS; ++i) m |= (s_prs[i] > 0 ? 1u : 0u) << i;
    s_mask = m;
  }

  // ---- per-wave (class) cost computation; lanes 0..26 are queries --------
  const int  c      = wid;
  const int  q      = lane;
  const int  qc     = (q < QPC) ? q : (QPC - 1);
  const int  j      = c * QPC + qc;
  const bool active = (q < QPC);

  const float x       = logits[(size_t)b * NQ + j];
  const float c_class = -1.0f / (1.0f + expf(-x));

  const float* ap = anchors + (size_t)j * 6;
  const float a0 = ap[0], a1 = ap[1], a2 = ap[2], a3 = ap[3], a4 = ap[4], a5 = ap[5];
  const float* tp = &s_tgt[c * 6];
  const float t0 = tp[0], t1 = tp[1], t2 = tp[2], t3 = tp[3], t4 = tp[4], t5 = tp[5];

  const float c_bbox = fabsf(a0 - t0) + fabsf(a1 - t1) + fabsf(a2 - t2) +
                       fabsf(a3 - t3) + fabsf(a4 - t4) + fabsf(a5 - t5);
  const float d0 = a0 - t0, d1 = a1 - t1, d2 = a2 - t2;
  const float c_center = sqrtf(d0 * d0 + d1 * d1 + d2 * d2);

  // GIoU on clipped anchor box vs target box (both cxcyczwhd -> xyzxyz)
  const float k0 = fmaxf(a0, 0.f), k1 = fmaxf(a1, 0.f), k2 = fmaxf(a2, 0.f);
  const float k3 = fmaxf(a3, 0.f), k4 = fmaxf(a4, 0.f), k5 = fmaxf(a5, 0.f);
  const float p1l0 = k0 - 0.5f * k3, p1l1 = k1 - 0.5f * k4, p1l2 = k2 - 0.5f * k5;
  const float p1h0 = k0 + 0.5f * k3, p1h1 = k1 + 0.5f * k4, p1h2 = k2 + 0.5f * k5;
  const float p2l0 = t0 - 0.5f * t3, p2l1 = t1 - 0.5f * t4, p2l2 = t2 - 0.5f * t5;
  const float p2h0 = t0 + 0.5f * t3, p2h1 = t1 + 0.5f * t4, p2h2 = t2 + 0.5f * t5;

  const float vol1 = (p1h0 - p1l0) * (p1h1 - p1l1) * (p1h2 - p1l2);
  const float vol2 = (p2h0 - p2l0) * (p2h1 - p2l1) * (p2h2 - p2l2);
  const float i0 = fmaxf(fminf(p1h0, p2h0) - fmaxf(p1l0, p2l0), 0.f);
  const float i1 = fmaxf(fminf(p1h1, p2h1) - fmaxf(p1l1, p2l1), 0.f);
  const float i2 = fmaxf(fminf(p1h2, p2h2) - fmaxf(p1l2, p2l2), 0.f);
  const float inter = i0 * i1 * i2;
  const float uni   = vol1 + vol2 - inter;
  const float iou   = inter / uni;
  const float e0 = fmaxf(fmaxf(p1h0, p2h0) - fminf(p1l0, p2l0), 0.f);
  const float e1 = fmaxf(fmaxf(p1h1, p2h1) - fminf(p1l1, p2l1), 0.f);
  const float e2 = fmaxf(fmaxf(p1h2, p2h2) - fminf(p1l2, p2l2), 0.f);
  const float volc = e0 * e1 * e2;
  const float cg = -(iou - (volc - uni) / volc);   // c_giou

  const float C = c_bbox + c_class + cg + c_center;

  // ---- top-k (k smallest C, first-index tie-break) via wave32 butterfly --
  int k = ntq[0];
  if (k < 1) k = 1;
  if (k > QPC) k = QPC;

  float Cv  = active ? C : INF_F;
  int   sel = 0;
  for (int it = 0; it < k; ++it) {
    float bc = Cv;
    int   bq = q;
#pragma unroll
    for (int off = 16; off > 0; off >>= 1) {
      const float oc = __shfl_xor(bc, off);
      const int   oq = __shfl_xor(bq, off);
      if (oc < bc || (oc == bc && oq < bq)) { bc = oc; bq = oq; }
    }
    if (q == bq) { sel = 1; Cv = INF_F; }
  }

  // ---- per-class min/max of c_giou -> soft_vals --------------------------
  float mx = active ? cg : -INF_F;
  float mn = active ? cg :  INF_F;
#pragma unroll
  for (int off = 16; off > 0; off >>= 1) {
    mx = fmaxf(mx, __shfl_xor(mx, off));
    mn = fminf(mn, __shfl_xor(mn, off));
  }
  const float sv = fmaxf((cg - mx) / (mn - mx), 0.f);

  const int pres = (s_prs[c] > 0) ? 1 : 0;
  if (active) {
    s_sv[j] = sv;
    matches[(size_t)b * NQ + j] = sel & pres;
  }
  __syncthreads();

  // ---- soft: closed form of the reference's sequential overlap loop ------
  if (tid < NQ) {
    const int jj = tid;
    const int ca = jj / QPC;
    const unsigned m = s_mask;
    int lo = jj - 26; if (lo < 0) lo = 0;
    const int hi = (jj < NCLS - 1) ? jj : (NCLS - 1);
    unsigned range = 0u;
    if (lo <= hi) range = ((1u << (hi + 1)) - 1u) & ~((1u << lo) - 1u);
    const unsigned bbits = (~m) & 0xFFFFFu & range;           // absent classes in B-range
    const int bestB = bbits ? (31 - __clz((int)bbits)) : -1;  // latest -1.0 writer
    const int bestA = ((m >> ca) & 1u) ? ca : -1;             // soft_vals writer
    float v = 0.0f;
    if (bestB > bestA)      v = -1.0f;
    else if (bestA >= 0)    v = s_sv[jj];
    soft[(size_t)b * NQ + jj] = v;
  }
}

extern "C" void kernel_launch(void* const* d_in, const int* in_sizes, int n_in,
                              void* d_out, int out_size, void* d_ws, size_t ws_size,
                              hipStream_t stream) {
  const float* logits  = (const float*)d_in[0];
  // d_in[1] = pred_boxes is unused by the reference (anchors are broadcast).
  const float* anchors = (const float*)d_in[2];
  const float* tgtb    = (const float*)d_in[3];
  const int*   tgtp    = (const int*)d_in[4];
  const int*   ntq     = (const int*)d_in[5];

  const int bs = in_sizes[0] / NQ;

  int*   matches = (int*)d_out;
  float* soft    = (float*)d_out + (size_t)bs * NQ;

  matcher_kernel<<<bs, 640, 0, stream>>>(logits, anchors, tgtb, tgtp, ntq,
                                         matches, soft);
}